// GATNet_81415400063708
// MI455X (gfx1250) — compile-verified
//
#include <hip/hip_runtime.h>
#include <hip/hip_bf16.h>

// ---------------------------------------------------------------------------
// 2-layer GAT (PyG GATConv semantics) for MI455X / gfx1250.
// Layer1 GEMM uses V_WMMA_F32_16X16X4_F32 (full fp32 precision).
// Edge softmax uses order-preserving uint encoding + atomicMax (no CAS loop),
// aggregation uses native f32 atomic adds (unsafeAtomicAdd ->
// global_atomic_add_f32). Everything f32 end-to-end. Edge gathers use float4.
// ---------------------------------------------------------------------------

typedef __attribute__((ext_vector_type(2))) float v2f;
typedef __attribute__((ext_vector_type(8))) float v8f;

#define NEG_SLOPE 0.2f
#define ENC_NEG_INF 0x007FFFFFu  // encf(-inf)

__device__ __forceinline__ unsigned encf(float f) {
    unsigned u = __float_as_uint(f);
    return (u & 0x80000000u) ? ~u : (u | 0x80000000u);
}
__device__ __forceinline__ float decf(unsigned u) {
    unsigned v = (u & 0x80000000u) ? (u & 0x7FFFFFFFu) : ~u;
    return __uint_as_float(v);
}
__device__ __forceinline__ float lrelu(float x) { return x > 0.0f ? x : NEG_SLOPE * x; }

__device__ __forceinline__ void fatomic_add(float* p, float v) {
    unsafeAtomicAdd(p, v);                         // -> global_atomic_add_f32
}

__device__ __forceinline__ void edge_sd(const long long* __restrict__ ei, int E,
                                        int eid, int& s, int& d) {
    if (eid < E) { s = (int)ei[eid]; d = (int)ei[(size_t)E + eid]; }
    else         { s = d = eid - E; }              // self loops appended
}

// --------------------------- layer 1 GEMM (WMMA) ---------------------------
// h1[N,32] = x[N,2] @ W1[2,32], K padded 2->4 with zeros.
// One wave: 16-node M-tile, two 16-col N-tiles -> 2x v_wmma_f32_16x16x4_f32.
__global__ void k_gemm1_wmma(const float* __restrict__ x, const float* __restrict__ W1,
                             float* __restrict__ h1, int nTiles, int N) {
    const int lane = threadIdx.x & 31;
    const int tile = blockIdx.x * (blockDim.x >> 5) + (threadIdx.x >> 5);
    if (tile >= nTiles) return;                    // wave-uniform: EXEC stays all-1s
    const int m  = lane & 15;
    const bool hi = lane >= 16;                    // hi lanes carry K=2,3 (zero pad)
    const int node = tile * 16 + m;
    const bool fullTile = (tile * 16 + 16 <= N);   // wave-uniform

    // Branchless A load: clamp address, select-zero pad lanes (no EXEC dance).
    const int nc = node < N ? node : (N - 1);
    const float x0 = x[nc * 2 + 0];
    const float x1 = x[nc * 2 + 1];
    const bool zA = hi || (node >= N);
    v2f A;
    A.x = zA ? 0.0f : x0;                          // K=0
    A.y = zA ? 0.0f : x1;                          // K=1

#pragma unroll
    for (int t = 0; t < 2; ++t) {
        const int col = t * 16 + m;
        const float w0 = W1[col];                  // K=0 row of W1
        const float w1 = W1[32 + col];             // K=1 row of W1
        v2f B;
        B.x = hi ? 0.0f : w0;
        B.y = hi ? 0.0f : w1;
        v8f C = {};
        C = __builtin_amdgcn_wmma_f32_16x16x4_f32(false, A, false, B,
                                                  (short)0, C, false, false);
        // D layout: VGPR r, lanes 0-15 -> M=r, lanes 16-31 -> M=r+8; col = lane&15.
        const int row0 = tile * 16 + (hi ? 8 : 0);
        float* p = h1 + (size_t)row0 * 32 + col;
        if (fullTile) {
#pragma unroll
            for (int r = 0; r < 8; ++r) p[r * 32] = C[r];  // const-offset stores
        } else {
#pragma unroll
            for (int r = 0; r < 8; ++r)
                if (row0 + r < N) p[r * 32] = C[r];
        }
    }
}

// ------------------------- init accumulators/maxima ------------------------
__global__ void k_init(float* __restrict__ out1, float* __restrict__ s1,
                       unsigned* __restrict__ m1e, float* __restrict__ o2a,
                       float* __restrict__ s2, unsigned* __restrict__ m2e, int N) {
    const int i = blockIdx.x * blockDim.x + threadIdx.x;
    if (i < N * 32) out1[i] = 0.0f;
    if (i < N * 4)  { s1[i] = 0.0f; m1e[i] = ENC_NEG_INF; }
    if (i < N * 2)  o2a[i] = 0.0f;
    if (i < N)      { s2[i] = 0.0f; m2e[i] = ENC_NEG_INF; }
}

// --------------------- layer 1 per-node attention logits -------------------
__global__ void k_logits1(const float* __restrict__ h1, const float* __restrict__ attS,
                          const float* __restrict__ attD, float* __restrict__ aS,
                          float* __restrict__ aD, int N) {
    const int n = blockIdx.x * blockDim.x + threadIdx.x;
    if (n >= N) return;
    const float4* hp = (const float4*)(h1 + (size_t)n * 32);
    float4 as4, ad4;
    float* asp = (float*)&as4;
    float* adp = (float*)&ad4;
#pragma unroll
    for (int h = 0; h < 4; ++h) {
        const float4 v0 = hp[h * 2 + 0];
        const float4 v1 = hp[h * 2 + 1];
        float as = v0.x * attS[h * 8 + 0] + v0.y * attS[h * 8 + 1] +
                   v0.z * attS[h * 8 + 2] + v0.w * attS[h * 8 + 3] +
                   v1.x * attS[h * 8 + 4] + v1.y * attS[h * 8 + 5] +
                   v1.z * attS[h * 8 + 6] + v1.w * attS[h * 8 + 7];
        float ad = v0.x * attD[h * 8 + 0] + v0.y * attD[h * 8 + 1] +
                   v0.z * attD[h * 8 + 2] + v0.w * attD[h * 8 + 3] +
                   v1.x * attD[h * 8 + 4] + v1.y * attD[h * 8 + 5] +
                   v1.z * attD[h * 8 + 6] + v1.w * attD[h * 8 + 7];
        asp[h] = as;
        adp[h] = ad;
    }
    ((float4*)aS)[n] = as4;
    ((float4*)aD)[n] = ad4;
}

// ------------------------------ edge passes L1 -----------------------------
__global__ void k_edge_max1(const long long* __restrict__ ei, int E, int Etot,
                            const float* __restrict__ aS, const float* __restrict__ aD,
                            unsigned* __restrict__ m1e) {
    const int eid = blockIdx.x * blockDim.x + threadIdx.x;
    if (eid >= Etot) return;
    int s, d; edge_sd(ei, E, eid, s, d);
    const float4 as4 = ((const float4*)aS)[s];
    const float4 ad4 = ((const float4*)aD)[d];
    const float* a = (const float*)&as4;
    const float* b = (const float*)&ad4;
#pragma unroll
    for (int h = 0; h < 4; ++h)
        atomicMax(&m1e[d * 4 + h], encf(lrelu(a[h] + b[h])));
}

__global__ void k_edge_expsum1(const long long* __restrict__ ei, int E, int Etot,
                               const float* __restrict__ aS, const float* __restrict__ aD,
                               const unsigned* __restrict__ m1e, float* __restrict__ s1,
                               float* __restrict__ alpha1) {
    const int eid = blockIdx.x * blockDim.x + threadIdx.x;
    if (eid >= Etot) return;
    int s, d; edge_sd(ei, E, eid, s, d);
    const float4 as4 = ((const float4*)aS)[s];
    const float4 ad4 = ((const float4*)aD)[d];
    const uint4  me4 = ((const uint4*)m1e)[d];
    const float*    a = (const float*)&as4;
    const float*    b = (const float*)&ad4;
    const unsigned* me = (const unsigned*)&me4;
    float4 ee4;
    float* ee = (float*)&ee4;
#pragma unroll
    for (int h = 0; h < 4; ++h) {
        ee[h] = expf(lrelu(a[h] + b[h]) - decf(me[h]));
        fatomic_add(&s1[d * 4 + h], ee[h]);
    }
    ((float4*)alpha1)[eid] = ee4;                  // stage ee in output region
}

__global__ void k_edge_agg1(const long long* __restrict__ ei, int E, int Etot,
                            const float* __restrict__ s1, const float* __restrict__ h1,
                            float* __restrict__ alpha1, float* __restrict__ out1) {
    const int eid = blockIdx.x * blockDim.x + threadIdx.x;
    if (eid >= Etot) return;
    int s, d; edge_sd(ei, E, eid, s, d);
    const float4 ee4 = ((const float4*)alpha1)[eid];
    const float4 ss4 = ((const float4*)s1)[d];
    const float* ee = (const float*)&ee4;
    const float* sv = (const float*)&ss4;
    float4 al4;
    float* al = (float*)&al4;
#pragma unroll
    for (int h = 0; h < 4; ++h) al[h] = ee[h] / (sv[h] + 1e-16f);
    ((float4*)alpha1)[eid] = al4;                  // final alpha1 output
    const float4* hrow = (const float4*)(h1 + (size_t)s * 32);
    float* orow = out1 + (size_t)d * 32;
#pragma unroll
    for (int h = 0; h < 4; ++h) {
        const float4 v0 = hrow[h * 2 + 0];
        const float4 v1 = hrow[h * 2 + 1];
        const float  a  = al[h];
        fatomic_add(&orow[h * 8 + 0], v0.x * a);
        fatomic_add(&orow[h * 8 + 1], v0.y * a);
        fatomic_add(&orow[h * 8 + 2], v0.z * a);
        fatomic_add(&orow[h * 8 + 3], v0.w * a);
        fatomic_add(&orow[h * 8 + 4], v1.x * a);
        fatomic_add(&orow[h * 8 + 5], v1.y * a);
        fatomic_add(&orow[h * 8 + 6], v1.z * a);
        fatomic_add(&orow[h * 8 + 7], v1.w * a);
    }
}

// -------- layer 2 node stage: bias + ELU + GEMM(32->2) + logits ------------
__global__ void k_node2(const float* __restrict__ out1, const float* __restrict__ b1,
                        const float* __restrict__ W2, const float* __restrict__ attS,
                        const float* __restrict__ attD, float* __restrict__ g2,
                        float* __restrict__ aS, float* __restrict__ aD, int N) {
    const int n = blockIdx.x * blockDim.x + threadIdx.x;
    if (n >= N) return;
    float g0 = 0.0f, g1 = 0.0f;
#pragma unroll
    for (int k = 0; k < 32; ++k) {
        float v = out1[(size_t)n * 32 + k] + b1[k];
        v = v > 0.0f ? v : expm1f(v);              // ELU(alpha=1)
        g0 += v * W2[k * 2 + 0];
        g1 += v * W2[k * 2 + 1];
    }
    g2[n * 2 + 0] = g0;
    g2[n * 2 + 1] = g1;
    aS[n] = g0 * attS[0] + g1 * attS[1];
    aD[n] = g0 * attD[0] + g1 * attD[1];
}

// ------------------------------ edge passes L2 -----------------------------
__global__ void k_edge_max2(const long long* __restrict__ ei, int E, int Etot,
                            const float* __restrict__ aS, const float* __restrict__ aD,
                            unsigned* __restrict__ m2e) {
    const int eid = blockIdx.x * blockDim.x + threadIdx.x;
    if (eid >= Etot) return;
    int s, d; edge_sd(ei, E, eid, s, d);
    atomicMax(&m2e[d], encf(lrelu(aS[s] + aD[d])));
}

__global__ void k_edge_expsum2(const long long* __restrict__ ei, int E, int Etot,
                               const float* __restrict__ aS, const float* __restrict__ aD,
                               const unsigned* __restrict__ m2e, float* __restrict__ s2,
                               float* __restrict__ alpha2) {
    const int eid = blockIdx.x * blockDim.x + threadIdx.x;
    if (eid >= Etot) return;
    int s, d; edge_sd(ei, E, eid, s, d);
    const float ee = expf(lrelu(aS[s] + aD[d]) - decf(m2e[d]));
    alpha2[eid] = ee;
    fatomic_add(&s2[d], ee);
}

__global__ void k_edge_agg2(const long long* __restrict__ ei, int E, int Etot,
                            const float* __restrict__ s2, const float* __restrict__ g2,
                            float* __restrict__ alpha2, float* __restrict__ o2a) {
    const int eid = blockIdx.x * blockDim.x + threadIdx.x;
    if (eid >= Etot) return;
    int s, d; edge_sd(ei, E, eid, s, d);
    const float a = alpha2[eid] / (s2[d] + 1e-16f);
    alpha2[eid] = a;                               // final alpha2 output
    const float2 g = ((const float2*)g2)[s];
    fatomic_add(&o2a[d * 2 + 0], g.x * a);
    fatomic_add(&o2a[d * 2 + 1], g.y * a);
}

// ------------------------------ final output -------------------------------
__global__ void k_final(const float* __restrict__ o2a, const float* __restrict__ b2,
                        float* __restrict__ out, int N) {
    const int n = blockIdx.x * blockDim.x + threadIdx.x;
    if (n >= N) return;
    float2 r = ((const float2*)o2a)[n];
    r.x += b2[0];                                  // heads=1 -> mean is identity
    r.y += b2[1];
    ((float2*)out)[n] = r;
}

// ---------------------------------------------------------------------------
extern "C" void kernel_launch(void* const* d_in, const int* in_sizes, int n_in,
                              void* d_out, int out_size, void* d_ws, size_t ws_size,
                              hipStream_t stream) {
    const float*     x    = (const float*)d_in[0];
    const long long* ei   = (const long long*)d_in[1];   // int64 [2,E]
    // d_in[2] = edge_attr : ignored (GATConv built without edge_dim)
    const float*     W1   = (const float*)d_in[3];
    const float*     atS1 = (const float*)d_in[4];
    const float*     atD1 = (const float*)d_in[5];
    const float*     b1   = (const float*)d_in[6];
    const float*     W2   = (const float*)d_in[7];
    const float*     atS2 = (const float*)d_in[8];
    const float*     atD2 = (const float*)d_in[9];
    const float*     b2   = (const float*)d_in[10];

    const int N    = in_sizes[0] / 2;
    const int E    = in_sizes[1] / 2;
    const int Etot = E + N;                        // with self loops

    // workspace carve-up (all 4-byte elements; 16B alignment maintained)
    float* ws = (float*)d_ws;
    float*    h1   = ws;  ws += (size_t)N * 32;
    float*    aS1  = ws;  ws += (size_t)N * 4;
    float*    aD1  = ws;  ws += (size_t)N * 4;
    unsigned* m1e  = (unsigned*)ws; ws += (size_t)N * 4;
    float*    s1   = ws;  ws += (size_t)N * 4;
    float*    out1 = ws;  ws += (size_t)N * 32;
    float*    g2   = ws;  ws += (size_t)N * 2;
    float*    aS2  = ws;  ws += (size_t)N;
    float*    aD2  = ws;  ws += (size_t)N;
    unsigned* m2e  = (unsigned*)ws; ws += (size_t)N;
    float*    s2   = ws;  ws += (size_t)N;
    float*    o2a  = ws;  ws += (size_t)N * 2;

    // d_out = [out (N*2) | alpha1 (Etot*4) | alpha2 (Etot)]
    float* out    = (float*)d_out;
    float* alpha1 = out + (size_t)N * 2;
    float* alpha2 = alpha1 + (size_t)Etot * 4;

    const int B = 256;
    const int nodeGrid = (N + B - 1) / B;
    const int edgeGrid = (Etot + B - 1) / B;
    const int initGrid = (N * 32 + B - 1) / B;
    const int nTiles   = (N + 15) / 16;
    const int gemmGrid = (nTiles + 7) / 8;         // 8 waves (16-node tiles) per block

    k_init<<<initGrid, B, 0, stream>>>(out1, s1, m1e, o2a, s2, m2e, N);
    k_gemm1_wmma<<<gemmGrid, B, 0, stream>>>(x, W1, h1, nTiles, N);
    k_logits1<<<nodeGrid, B, 0, stream>>>(h1, atS1, atD1, aS1, aD1, N);

    k_edge_max1<<<edgeGrid, B, 0, stream>>>(ei, E, Etot, aS1, aD1, m1e);
    k_edge_expsum1<<<edgeGrid, B, 0, stream>>>(ei, E, Etot, aS1, aD1, m1e, s1, alpha1);
    k_edge_agg1<<<edgeGrid, B, 0, stream>>>(ei, E, Etot, s1, h1, alpha1, out1);

    k_node2<<<nodeGrid, B, 0, stream>>>(out1, b1, W2, atS2, atD2, g2, aS2, aD2, N);

    k_edge_max2<<<edgeGrid, B, 0, stream>>>(ei, E, Etot, aS2, aD2, m2e);
    k_edge_expsum2<<<edgeGrid, B, 0, stream>>>(ei, E, Etot, aS2, aD2, m2e, s2, alpha2);
    k_edge_agg2<<<edgeGrid, B, 0, stream>>>(ei, E, Etot, s2, g2, alpha2, o2a);

    k_final<<<nodeGrid, B, 0, stream>>>(o2a, b2, out, N);
}